// SuperpointGraphModule_7146825581108
// MI455X (gfx1250) — compile-verified
//
#include <hip/hip_runtime.h>
#include <math.h>

typedef __attribute__((ext_vector_type(2))) float v2f;
typedef __attribute__((ext_vector_type(8))) float v8f;

#define EPSW 1e-8f
#define LNEPS 1e-5f
#define BIGD 3.4e38f

__device__ __forceinline__ float wave_sum(float v) {
#pragma unroll
    for (int off = 16; off >= 1; off >>= 1) v += __shfl_xor(v, off, 32);
    return v;
}

// ---------------------------------------------------------------------------
// K0: pad coords to [N,4] (w=0) and compute squared norms
// ---------------------------------------------------------------------------
__global__ void prep_kernel(const float* __restrict__ crd, float* __restrict__ c4,
                            float* __restrict__ sq, int n) {
    int i = blockIdx.x * blockDim.x + threadIdx.x;
    if (i >= n) return;
    float x = crd[3 * i], y = crd[3 * i + 1], z = crd[3 * i + 2];
    float4 o; o.x = x; o.y = y; o.z = z; o.w = 0.0f;
    ((float4*)c4)[i] = o;
    sq[i] = x * x + y * y + z * z;
}

// ---------------------------------------------------------------------------
// K1: fused  LN1 -> (x+x) -> LN2  ; also row L2 norms (for cosine sim).
// One wave32 per row of 64; each lane owns 2 columns.
// ---------------------------------------------------------------------------
__global__ __launch_bounds__(256) void ln2_kernel(
    const float* __restrict__ in, const float* __restrict__ g1, const float* __restrict__ b1,
    const float* __restrict__ g2, const float* __restrict__ b2,
    float* __restrict__ xout, float* __restrict__ rn, int n) {
    int row = blockIdx.x * (blockDim.x >> 5) + (threadIdx.x >> 5);
    if (row >= n) return;
    int lane = threadIdx.x & 31;
    float2 v = ((const float2*)(in + (size_t)row * 64))[lane];
    float s  = wave_sum(v.x + v.y);
    float ss = wave_sum(v.x * v.x + v.y * v.y);
    float m  = s * (1.0f / 64.0f);
    float var = ss * (1.0f / 64.0f) - m * m;
    float r  = rsqrtf(var + LNEPS);
    float y0 = (v.x - m) * r * g1[2 * lane]     + b1[2 * lane];
    float y1 = (v.y - m) * r * g1[2 * lane + 1] + b1[2 * lane + 1];
    y0 *= 2.0f; y1 *= 2.0f;                       // x = x + x
    s  = wave_sum(y0 + y1);
    ss = wave_sum(y0 * y0 + y1 * y1);
    m  = s * (1.0f / 64.0f);
    var = ss * (1.0f / 64.0f) - m * m;
    r  = rsqrtf(var + LNEPS);
    float z0 = (y0 - m) * r * g2[2 * lane]     + b2[2 * lane];
    float z1 = (y1 - m) * r * g2[2 * lane + 1] + b2[2 * lane + 1];
    float2 zo; zo.x = z0; zo.y = z1;
    ((float2*)(xout + (size_t)row * 64))[lane] = zo;
    float nr = wave_sum(z0 * z0 + z1 * z1);
    if (lane == 0) rn[row] = fmaxf(sqrtf(nr), EPSW);
}

// ---------------------------------------------------------------------------
// K2: brute-force KNN (k=3) via WMMA f32 16x16x4 Gram tiles.
// One wave per 16 query rows; loops over 16-candidate tiles; per-lane top-3
// per C-row-slot kept in VGPRs; LDS merge at the end.
// d = |q|^2 + |c|^2 - 2*q.c  (identical formula to the reference)
// ---------------------------------------------------------------------------
__global__ __launch_bounds__(32) void knn_kernel(
    const float* __restrict__ c4, const float* __restrict__ sq,
    int* __restrict__ knn, int n) {
    int lane  = threadIdx.x;
    int qbase = blockIdx.x * 16;
    int half  = lane >> 4;      // K-half for A/B fragments
    int nl    = lane & 15;
    v2f a;
    {   // A fragment: A[m=nl][K = 2*half + v]
        const float* p = c4 + (size_t)(qbase + nl) * 4 + half * 2;
        a[0] = p[0]; a[1] = p[1];
    }
    float sqm[8];
#pragma unroll
    for (int r2 = 0; r2 < 8; ++r2) sqm[r2] = sq[qbase + half * 8 + r2];

    float bd[8][3]; int bi[8][3];
#pragma unroll
    for (int r2 = 0; r2 < 8; ++r2) {
        bd[r2][0] = bd[r2][1] = bd[r2][2] = BIGD;
        bi[r2][0] = bi[r2][1] = bi[r2][2] = 0;
    }
    int ntiles = n >> 4;
    for (int t = 0; t < ntiles; ++t) {
        int j = t * 16 + nl;
        v2f b;
        {   // B fragment: B[K = 2*half + v][n = nl]
            const float* p = c4 + (size_t)j * 4 + half * 2;
            b[0] = p[0]; b[1] = p[1];
        }
        float sqj = sq[j];
        v8f c = {};
        c = __builtin_amdgcn_wmma_f32_16x16x4_f32(false, a, false, b, (short)0, c,
                                                  false, false);
#pragma unroll
        for (int r2 = 0; r2 < 8; ++r2) {
            int gm = qbase + half * 8 + r2;     // global query row for this slot
            float d = sqm[r2] + sqj - 2.0f * c[r2];
            if (j == gm) d = BIGD;              // exclude self
            if (d < bd[r2][2]) {
                if (d < bd[r2][1]) {
                    bd[r2][2] = bd[r2][1]; bi[r2][2] = bi[r2][1];
                    if (d < bd[r2][0]) {
                        bd[r2][1] = bd[r2][0]; bi[r2][1] = bi[r2][0];
                        bd[r2][0] = d; bi[r2][0] = j;
                    } else { bd[r2][1] = d; bi[r2][1] = j; }
                } else { bd[r2][2] = d; bi[r2][2] = j; }
            }
        }
    }
    __shared__ float sd[32 * 24];
    __shared__ int   si[32 * 24];
#pragma unroll
    for (int r2 = 0; r2 < 8; ++r2)
#pragma unroll
        for (int k = 0; k < 3; ++k) {
            sd[lane * 24 + r2 * 3 + k] = bd[r2][k];
            si[lane * 24 + r2 * 3 + k] = bi[r2][k];
        }
    __syncthreads();
    if (lane < 16) {            // lane m merges row m's 16 lane-local top-3s
        int h = lane >> 3, r2 = lane & 7, lo = h * 16;
        float d0 = BIGD, d1 = BIGD, d2 = BIGD;
        int i0 = 0, i1 = 0, i2 = 0;
        for (int l2 = 0; l2 < 16; ++l2)
#pragma unroll
            for (int k = 0; k < 3; ++k) {
                float d = sd[(lo + l2) * 24 + r2 * 3 + k];
                int  jj = si[(lo + l2) * 24 + r2 * 3 + k];
                if (d < d2) {
                    if (d < d1) {
                        d2 = d1; i2 = i1;
                        if (d < d0) { d1 = d0; i1 = i0; d0 = d; i0 = jj; }
                        else        { d1 = d;  i1 = jj; }
                    } else { d2 = d; i2 = jj; }
                }
            }
        int q = qbase + lane;
        knn[q * 3 + 0] = i0; knn[q * 3 + 1] = i1; knn[q * 3 + 2] = i2;
    }
}

// ---------------------------------------------------------------------------
// K3: edge weights  ew[e] = sigmoid(cos(x[src], x[dst])), one wave per edge.
// ---------------------------------------------------------------------------
__global__ __launch_bounds__(256) void edge_kernel(
    const float* __restrict__ x, const float* __restrict__ rn,
    const int* __restrict__ knn, float* __restrict__ ew, int nedges) {
    int e = blockIdx.x * (blockDim.x >> 5) + (threadIdx.x >> 5);
    if (e >= nedges) return;
    int lane = threadIdx.x & 31;
    int dst = e / 3;
    int src = knn[e];
    float2 av = ((const float2*)(x + (size_t)src * 64))[lane];
    float2 bv = ((const float2*)(x + (size_t)dst * 64))[lane];
    float dot = wave_sum(av.x * bv.x + av.y * bv.y);
    if (lane == 0) {
        float cs = dot / (rn[src] * rn[dst]);
        ew[e] = 1.0f / (1.0f + expf(-cs));
    }
}

// K3b: deg[i] = 1 (self loop) + sum of its 3 in-edge weights (dst(e)=e/3)
__global__ void deg_kernel(const float* __restrict__ ew, float* __restrict__ deg, int n) {
    int i = blockIdx.x * blockDim.x + threadIdx.x;
    if (i >= n) return;
    deg[i] = 1.0f + ew[3 * i] + ew[3 * i + 1] + ew[3 * i + 2];
}

// ---------------------------------------------------------------------------
// K4/K6: O[m x 64] = A[m x 64] @ W[64 x 64] via WMMA f32 16x16x4, K-loop of 16.
// One wave per 16x16 output tile.
// ---------------------------------------------------------------------------
__global__ __launch_bounds__(256) void gemm64_kernel(
    const float* __restrict__ A, const float* __restrict__ W,
    float* __restrict__ O, int m) {
    int wid = blockIdx.x * (blockDim.x >> 5) + (threadIdx.x >> 5);
    int mt = wid >> 2, nt = wid & 3;
    if (mt * 16 >= m) return;
    int lane = threadIdx.x & 31;
    int half = lane >> 4, nl = lane & 15;
    const float* arow = A + (size_t)(mt * 16 + nl) * 64 + half * 2;   // A[m][2h+v]
    const float* wcol = W + (size_t)(half * 2) * 64 + nt * 16 + nl;   // W[2h+v][n]
    v8f acc = {};
#pragma unroll
    for (int k0 = 0; k0 < 64; k0 += 4) {
        v2f a; a[0] = arow[k0];          a[1] = arow[k0 + 1];
        v2f b; b[0] = wcol[k0 * 64];     b[1] = wcol[k0 * 64 + 64];
        acc = __builtin_amdgcn_wmma_f32_16x16x4_f32(false, a, false, b, (short)0, acc,
                                                    false, false);
    }
    float* orow = O + (size_t)(mt * 16 + half * 8) * 64 + nt * 16 + nl;
#pragma unroll
    for (int r = 0; r < 8; ++r) orow[(size_t)r * 64] = acc[r];
}

// ---------------------------------------------------------------------------
// K5: GCN aggregate (gather form) + bias + LayerNorm + ReLU -> G
// out1[i] = b1 + di*( sum_k d_sk*ew_k*H[s_k] + di*H[i] ) ; G = relu(LN(out1))
// ---------------------------------------------------------------------------
__global__ __launch_bounds__(256) void agg_ln_relu_kernel(
    const float* __restrict__ H, const int* __restrict__ knn,
    const float* __restrict__ ew, const float* __restrict__ deg,
    const float* __restrict__ bias, const float* __restrict__ g,
    const float* __restrict__ bln, float* __restrict__ G, int n) {
    int i = blockIdx.x * (blockDim.x >> 5) + (threadIdx.x >> 5);
    if (i >= n) return;
    int lane = threadIdx.x & 31;
    int s0 = knn[3 * i], s1 = knn[3 * i + 1], s2 = knn[3 * i + 2];
    float di = rsqrtf(fmaxf(deg[i], EPSW));
    float c0 = di * rsqrtf(fmaxf(deg[s0], EPSW)) * ew[3 * i];
    float c1 = di * rsqrtf(fmaxf(deg[s1], EPSW)) * ew[3 * i + 1];
    float c2 = di * rsqrtf(fmaxf(deg[s2], EPSW)) * ew[3 * i + 2];
    float cs = di * di;
    float2 h0 = ((const float2*)(H + (size_t)s0 * 64))[lane];
    float2 h1 = ((const float2*)(H + (size_t)s1 * 64))[lane];
    float2 h2 = ((const float2*)(H + (size_t)s2 * 64))[lane];
    float2 hs = ((const float2*)(H + (size_t)i  * 64))[lane];
    float v0 = c0 * h0.x + c1 * h1.x + c2 * h2.x + cs * hs.x + bias[2 * lane];
    float v1 = c0 * h0.y + c1 * h1.y + c2 * h2.y + cs * hs.y + bias[2 * lane + 1];
    float s  = wave_sum(v0 + v1);
    float ss = wave_sum(v0 * v0 + v1 * v1);
    float m  = s * (1.0f / 64.0f);
    float var = ss * (1.0f / 64.0f) - m * m;
    float r  = rsqrtf(var + LNEPS);
    float2 zo;
    zo.x = fmaxf((v0 - m) * r * g[2 * lane]     + bln[2 * lane],     0.0f);
    zo.y = fmaxf((v1 - m) * r * g[2 * lane + 1] + bln[2 * lane + 1], 0.0f);
    ((float2*)(G + (size_t)i * 64))[lane] = zo;
}

// ---------------------------------------------------------------------------
// K7: second aggregate + residuals:  out = 2*x + b2 + agg(H2)
// (shortcut + (gcn2_out + x) = 2x + agg2 + b2)
// ---------------------------------------------------------------------------
__global__ __launch_bounds__(256) void agg_final_kernel(
    const float* __restrict__ H, const int* __restrict__ knn,
    const float* __restrict__ ew, const float* __restrict__ deg,
    const float* __restrict__ bias, const float* __restrict__ x,
    float* __restrict__ out, int n) {
    int i = blockIdx.x * (blockDim.x >> 5) + (threadIdx.x >> 5);
    if (i >= n) return;
    int lane = threadIdx.x & 31;
    int s0 = knn[3 * i], s1 = knn[3 * i + 1], s2 = knn[3 * i + 2];
    float di = rsqrtf(fmaxf(deg[i], EPSW));
    float c0 = di * rsqrtf(fmaxf(deg[s0], EPSW)) * ew[3 * i];
    float c1 = di * rsqrtf(fmaxf(deg[s1], EPSW)) * ew[3 * i + 1];
    float c2 = di * rsqrtf(fmaxf(deg[s2], EPSW)) * ew[3 * i + 2];
    float cs = di * di;
    float2 h0 = ((const float2*)(H + (size_t)s0 * 64))[lane];
    float2 h1 = ((const float2*)(H + (size_t)s1 * 64))[lane];
    float2 h2 = ((const float2*)(H + (size_t)s2 * 64))[lane];
    float2 hs = ((const float2*)(H + (size_t)i  * 64))[lane];
    float2 xv = ((const float2*)(x + (size_t)i  * 64))[lane];
    float2 o;
    o.x = 2.0f * xv.x + bias[2 * lane]     + c0 * h0.x + c1 * h1.x + c2 * h2.x + cs * hs.x;
    o.y = 2.0f * xv.y + bias[2 * lane + 1] + c0 * h0.y + c1 * h1.y + c2 * h2.y + cs * hs.y;
    ((float2*)(out + (size_t)i * 64))[lane] = o;
}

// ---------------------------------------------------------------------------
extern "C" void kernel_launch(void* const* d_in, const int* in_sizes, int n_in,
                              void* d_out, int out_size, void* d_ws, size_t ws_size,
                              hipStream_t stream) {
    const float* feat  = (const float*)d_in[0];   // [N,64]
    const float* crd   = (const float*)d_in[4];   // [N,3]
    const float* n1g   = (const float*)d_in[5];
    const float* n1b   = (const float*)d_in[6];
    const float* n2g   = (const float*)d_in[7];
    const float* n2b   = (const float*)d_in[8];
    const float* gng   = (const float*)d_in[9];
    const float* gnb   = (const float*)d_in[10];
    const float* W1    = (const float*)d_in[11];
    const float* b1    = (const float*)d_in[12];
    const float* W2    = (const float*)d_in[13];
    const float* b2    = (const float*)d_in[14];
    float* out = (float*)d_out;

    const int n = in_sizes[0] / 64;               // 16384
    const int ne = 3 * n;

    // workspace layout (floats):  ~205*N floats total (~13.4 MB at N=16K)
    float* x    = (float*)d_ws;                   // N*64
    float* rn   = x + (size_t)n * 64;             // N
    float* c4   = rn + n;                         // N*4
    float* sq   = c4 + (size_t)n * 4;             // N
    float* deg  = sq + n;                         // N
    float* ew   = deg + n;                        // 3N
    int*   knn  = (int*)(ew + (size_t)3 * n);     // 3N
    float* H    = (float*)(knn + (size_t)3 * n);  // N*64  (H1, then reused as H2)
    float* G    = H + (size_t)n * 64;             // N*64

    // K0: coord pad + squared norms
    prep_kernel<<<(n + 255) / 256, 256, 0, stream>>>(crd, c4, sq, n);
    // K1: double layernorm -> x, row norms
    ln2_kernel<<<(n + 7) / 8, 256, 0, stream>>>(feat, n1g, n1b, n2g, n2b, x, rn, n);
    // K2: WMMA-based brute-force KNN (k=3)
    knn_kernel<<<n / 16, 32, 0, stream>>>(c4, sq, knn, n);
    // K3: edge weights + degrees (atomic-free, deterministic)
    edge_kernel<<<(ne + 7) / 8, 256, 0, stream>>>(x, rn, knn, ew, ne);
    deg_kernel<<<(n + 255) / 256, 256, 0, stream>>>(ew, deg, n);
    // K4: H1 = x @ W1  (WMMA f32)
    gemm64_kernel<<<(n / 16) * 4 / 8, 256, 0, stream>>>(x, W1, H, n);
    // K5: aggregate + b1 + LN + ReLU -> G
    agg_ln_relu_kernel<<<(n + 7) / 8, 256, 0, stream>>>(H, knn, ew, deg, b1, gng, gnb, G, n);
    // K6: H2 = G @ W2  (WMMA f32, reuse H)
    gemm64_kernel<<<(n / 16) * 4 / 8, 256, 0, stream>>>(G, W2, H, n);
    // K7: out = 2x + b2 + aggregate(H2)
    agg_final_kernel<<<(n + 7) / 8, 256, 0, stream>>>(H, knn, ew, deg, b2, x, out, n);
}